// unit_gcn_22497038696548
// MI455X (gfx1250) — compile-verified
//
#include <hip/hip_runtime.h>
#include <hip/hip_bf16.h>

// Shapes: N=64, C=64, T=256, V=25, H=8, CO=64, BR=3
// Fused operator per head h:
//   M_h[(o,w),(c,v)] = sum_b w2[b,h,o,c] * w1[b,h,v,w]          (200x200, padded 208x224)
//   y[n, h*8+o, t, w] = relu( (M_h @ x)[...] * alpha + cadd + x_residual )
//
// ws layout (floats):
//   [0      .. 15000 )  w1  (BR*H*25*25)
//   [15360  .. 388096)  M   (8 * 208 * 224)
//   [388096 .. 388160)  alpha (64)
//   [388160 .. 388224)  cadd  (64)
// requires ws_size >= 1,552,896 bytes

typedef __attribute__((ext_vector_type(2))) float v2f;
typedef __attribute__((ext_vector_type(8))) float v8f;

// ---------------------------------------------------------------------------
// Prep 1: w1[b,h,v,w] = fc1/||fc1||_col + pos/||pos||_col  (norm over v, per w)
// ---------------------------------------------------------------------------
__global__ void gcn_prep_w1(const float* __restrict__ fc1,
                            const float* __restrict__ rpe,
                            const int* __restrict__ hops,
                            int rpeL,
                            float* __restrict__ w1) {
    __shared__ float sf[625];
    __shared__ float sp[625];
    __shared__ float nf[25];
    __shared__ float np_[25];
    int bh  = blockIdx.x;          // b*8 + h, 0..23
    int tid = threadIdx.x;
    const float* f  = fc1 + bh * 625;
    const float* rp = rpe + bh * rpeL;
    for (int i = tid; i < 625; i += 256) {
        sf[i] = f[i];
        sp[i] = rp[hops[i]];
    }
    __syncthreads();
    if (tid < 25) {
        float a = 0.f, b = 0.f;
        for (int v = 0; v < 25; ++v) {
            float u = sf[v * 25 + tid]; a += u * u;
            float p = sp[v * 25 + tid]; b += p * p;
        }
        nf[tid]  = sqrtf(a) + 1e-4f;
        np_[tid] = sqrtf(b) + 1e-4f;
    }
    __syncthreads();
    for (int i = tid; i < 625; i += 256) {
        int w = i % 25;
        w1[bh * 625 + i] = sf[i] / nf[w] + sp[i] / np_[w];
    }
}

// ---------------------------------------------------------------------------
// Prep 2: M[h][r=o*25+w (pad 208)][k=c*28+v (pad 224)] = sum_b w2[b,h,o,c]*w1[b,h,v,w]
// ---------------------------------------------------------------------------
__global__ void gcn_prep_M(const float* __restrict__ w2,
                           const float* __restrict__ w1,
                           float* __restrict__ M) {
    int idx = blockIdx.x * 256 + threadIdx.x;
    if (idx >= 8 * 208 * 224) return;
    int h  = idx / (208 * 224);
    int r  = (idx / 224) % 208;
    int k  = idx % 224;
    int c  = k / 28;
    int vv = k % 28;
    float m = 0.f;
    if (r < 200 && vv < 25) {
        int o = r / 25;
        int w = r - o * 25;
        #pragma unroll
        for (int b = 0; b < 3; ++b)
            m += w2[((b * 8 + h) * 8 + o) * 8 + c] * w1[(b * 8 + h) * 625 + vv * 25 + w];
    }
    M[idx] = m;
}

// ---------------------------------------------------------------------------
// Prep 3: fused BN/bias constants
// ---------------------------------------------------------------------------
__global__ void gcn_prep_c(const float* __restrict__ b2,
                           const float* __restrict__ gamma,
                           const float* __restrict__ beta,
                           const float* __restrict__ mean,
                           const float* __restrict__ var,
                           float* __restrict__ alpha,
                           float* __restrict__ cadd) {
    int co = threadIdx.x;
    if (co < 64) {
        float bsum = b2[co] + b2[64 + co] + b2[128 + co];
        float a = gamma[co] * rsqrtf(var[co] + 1e-5f);
        alpha[co] = a;
        cadd[co]  = (bsum - mean[co]) * a + beta[co];
    }
}

// ---------------------------------------------------------------------------
// Main: per block = (n, head, t-block of 32). LDS-staged x tile, f32 WMMA GEMM,
// fused epilogue (BN + bias + residual-from-LDS + relu), coalesced stores.
// Strategy: per mt tile, batch-load the 16x224 A panel into a register array
// (56 independent float2 loads, one wait), then stream B from LDS into
// back-to-back WMMAs with no global-load waits in the compute phase.
// ---------------------------------------------------------------------------
__global__ __launch_bounds__(256) void gcn_main(const float* __restrict__ x,
                                                const float* __restrict__ M,
                                                const float* __restrict__ alpha,
                                                const float* __restrict__ cadd,
                                                float* __restrict__ out) {
    __shared__ float ldsX[8 * 32 * 28];   // x tile, v padded 25->28, 28672 B
    __shared__ float ldsO[8 * 32 * 25];   // output tile, 25600 B

    int bid = blockIdx.x;
    int tb  = bid & 7;          // t-block 0..7
    int h   = (bid >> 3) & 7;   // head
    int n   = bid >> 6;         // batch
    int t0  = tb * 32;
    int tid = threadIdx.x;

    const float* xbase = x + (size_t)n * 409600 + (size_t)(h * 8) * 6400 + t0 * 25;

    // ---- stage x tile into LDS (coalesced; scatter into padded layout) ----
    for (int i = tid; i < 6400; i += 256) {
        int c = i / 800, r = i % 800;
        int t = r / 25,  v = r % 25;
        ldsX[c * 896 + t * 28 + v] = xbase[c * 6400 + r];
    }
    for (int i = tid; i < 768; i += 256) {  // zero v = 25..27 pads
        int c = i / 96, r = i % 96;
        int t = r / 3,  v = 25 + r % 3;
        ldsX[c * 896 + t * 28 + v] = 0.f;
    }
    __syncthreads();

    int lane = tid & 31;
    int wave = tid >> 5;
    int tc   = lane & 15;
    int hi2  = (lane & 16) ? 2 : 0;        // K lane offset for A/B fragments
    const float* Mh = M + h * 46592 + hi2; // 46592 = 208*224
    int ldo  = tc * 28 + hi2;

    for (int mt = wave; mt < 13; mt += 8) {
        // ---- batch-load A panel for this row tile into registers ----
        const float* Ap = Mh + (mt * 16 + tc) * 224;
        v2f areg[56];
        #pragma unroll
        for (int k = 0; k < 56; ++k)
            areg[k] = *(const v2f*)(Ap + 4 * k);

        v8f acc0 = {};
        v8f acc1 = {};
        #pragma unroll
        for (int c = 0; c < 8; ++c) {
            int lb = c * 896 + ldo;
            #pragma unroll
            for (int q = 0; q < 7; ++q) {
                v2f b0 = *(const v2f*)(&ldsX[lb + q * 4]);
                v2f b1 = *(const v2f*)(&ldsX[lb + 448 + q * 4]);   // cols +16
                acc0 = __builtin_amdgcn_wmma_f32_16x16x4_f32(
                          false, areg[c * 7 + q], false, b0, (short)0, acc0, false, false);
                acc1 = __builtin_amdgcn_wmma_f32_16x16x4_f32(
                          false, areg[c * 7 + q], false, b1, (short)0, acc1, false, false);
            }
        }
        // D layout: VGPR g -> row = mt*16 + g + (lane>=16 ? 8 : 0), col = tc
        int rbase = mt * 16 + ((lane & 16) ? 8 : 0);
        #pragma unroll
        for (int g = 0; g < 8; ++g) {
            int r = rbase + g;
            if (r < 200) {
                int o = r / 25;
                int w = r - o * 25;
                ldsO[o * 800 + tc * 25 + w]        = acc0[g];
                ldsO[o * 800 + (16 + tc) * 25 + w] = acc1[g];
            }
        }
    }
    __syncthreads();

    // ---- epilogue: BN + bias + residual (from LDS) + relu, coalesced store ----
    float* obase = out + (size_t)n * 409600 + (size_t)(h * 8) * 6400 + t0 * 25;
    for (int i = tid; i < 6400; i += 256) {
        int c = i / 800, r = i % 800;
        int t = r / 25,  v = r % 25;
        int co = h * 8 + c;
        float val = fmaf(ldsO[c * 800 + r], alpha[co], cadd[co]);
        val += ldsX[c * 896 + t * 28 + v];
        obase[c * 6400 + r] = fmaxf(val, 0.f);
    }
}

// ---------------------------------------------------------------------------
extern "C" void kernel_launch(void* const* d_in, const int* in_sizes, int n_in,
                              void* d_out, int out_size, void* d_ws, size_t ws_size,
                              hipStream_t stream) {
    const float* x     = (const float*)d_in[0];
    const float* fc1   = (const float*)d_in[1];
    const float* rpe   = (const float*)d_in[2];
    const float* w2    = (const float*)d_in[3];
    const float* b2    = (const float*)d_in[4];
    const float* gamma = (const float*)d_in[5];
    const float* beta  = (const float*)d_in[6];
    const float* mean  = (const float*)d_in[7];
    const float* var   = (const float*)d_in[8];
    const int*   hops  = (const int*)d_in[9];   // jax default int32

    int rpeL = in_sizes[2] / 24;                // BR*H = 24

    float* ws     = (float*)d_ws;
    float* w1     = ws;                 // 15000
    float* Mmat   = ws + 15360;         // 372736
    float* alpha  = ws + 388096;        // 64
    float* cadd   = ws + 388160;        // 64
    float* out    = (float*)d_out;

    gcn_prep_w1<<<24, 256, 0, stream>>>(fc1, rpe, hops, rpeL, w1);
    gcn_prep_c<<<1, 64, 0, stream>>>(b2, gamma, beta, mean, var, alpha, cadd);
    gcn_prep_M<<<(8 * 208 * 224 + 255) / 256, 256, 0, stream>>>(w2, w1, Mmat);
    gcn_main<<<4096, 256, 0, stream>>>(x, Mmat, alpha, cadd, out);
}